// SymmetricFaceConv_13099650253569
// MI455X (gfx1250) — compile-verified
//
#include <hip/hip_runtime.h>

typedef float v2f __attribute__((ext_vector_type(2)));
typedef float v8f __attribute__((ext_vector_type(8)));

#define C_IN            32
#define KTOT            96      // 3 * 32 concatenated reduction dim
#define NCHUNK          24      // KTOT / 4
#define LDS_STRIDE      100     // padded row stride in dwords (conflict-free for 16-lane groups)
#define FACES_PER_WAVE  16
#define WAVES_PER_BLOCK 8
#define FACES_PER_BLOCK (FACES_PER_WAVE * WAVES_PER_BLOCK)
#define LDS_W_OFF       0                         // W: [32][LDS_STRIDE]
#define LDS_G_OFF       (32 * LDS_STRIDE)         // G: [8 waves][16][LDS_STRIDE]
#define LDS_TOTAL       (LDS_G_OFF + WAVES_PER_BLOCK * FACES_PER_WAVE * LDS_STRIDE)

__global__ __launch_bounds__(256) void SymmetricFaceConv_kernel(
    const float* __restrict__ x,
    const float* __restrict__ w0,
    const float* __restrict__ w1,
    const float* __restrict__ w2,
    const float* __restrict__ bias,
    const int*   __restrict__ fn,
    float* __restrict__ out,
    int nfaces)
{
    __shared__ float lds[LDS_TOTAL];

    const int tid  = threadIdx.x;
    const int lane = tid & 31;
    const int wave = tid >> 5;

    // ---------------- Stage Wcat = [w0|w1|w2] into LDS: [o][96], stride 100 ----------
    for (int i = tid; i < 32 * KTOT; i += 256) {
        int o = i / KTOT;
        int c = i - o * KTOT;
        const float* wsrc = (c < 32) ? w0 : ((c < 64) ? w1 : w2);
        lds[LDS_W_OFF + o * LDS_STRIDE + c] = wsrc[o * 32 + (c & 31)];
    }
    __syncthreads();

    // ---------------- Gather phase: build G = [g0|g1|g2] (16 faces x 96) in LDS ------
    const int face_base = blockIdx.x * FACES_PER_BLOCK + wave * FACES_PER_WAVE;
    float* gbase = &lds[LDS_G_OFF + wave * (FACES_PER_WAVE * LDS_STRIDE)];

    const int q    = lane & 7;   // channel quad: channels 4q..4q+3
    const int fsub = lane >> 3;  // face-within-pass: 0..3

    for (int pass = 0; pass < 4; ++pass) {
        const int frel  = pass * 4 + fsub;
        int face        = face_base + frel;
        int cface       = face < (nfaces - 1) ? face : (nfaces - 1);
        const int* fnrow = fn + (size_t)cface * 9;

        float4 a0 = make_float4(0.f, 0.f, 0.f, 0.f);
        float4 a1 = a0, a2 = a0;
        #pragma unroll
        for (int j = 0; j < 9; ++j) {
            int idx = fnrow[j];
            float4 v = make_float4(0.f, 0.f, 0.f, 0.f);
            if (idx < nfaces) {  // rows >= nfaces are the zero padding
                v = *(const float4*)(x + (size_t)idx * C_IN + q * 4);
            }
            if (j == 0)      { a0 = v; }
            else if (j & 1)  { a1.x += v.x; a1.y += v.y; a1.z += v.z; a1.w += v.w; }
            else             { a2.x += v.x; a2.y += v.y; a2.z += v.z; a2.w += v.w; }
        }
        float* grow = gbase + frel * LDS_STRIDE + q * 4;
        *(float4*)(grow +  0) = a0;   // g0 -> channels  0..31
        *(float4*)(grow + 32) = a1;   // g1 -> channels 32..63
        *(float4*)(grow + 64) = a2;   // g2 -> channels 64..95
    }
    // G region is per-wave private; intra-wave DS ordering covers the RAW below.

    // ---------------- WMMA phase: D(16x32) = G(16x96) x Wcat^T(96x32) + bias ----------
    const int m = lane & 15;   // A: matrix row (face) / B,C: column (output channel)
    const int h = lane >> 4;   // K sub-pair selector within each K=4 chunk

    v8f acc0, acc1;
    {
        float b0v = bias[m];
        float b1v = bias[m + 16];
        #pragma unroll
        for (int r = 0; r < 8; ++r) { acc0[r] = b0v; acc1[r] = b1v; }
    }

    const float* arow  = gbase + m * LDS_STRIDE + 2 * h;
    const float* brow0 = &lds[LDS_W_OFF +  m        * LDS_STRIDE + 2 * h];
    const float* brow1 = &lds[LDS_W_OFF + (m + 16)  * LDS_STRIDE + 2 * h];

    #pragma unroll
    for (int k = 0; k < NCHUNK; ++k) {
        v2f a  = *(const v2f*)(arow  + 4 * k);  // A[m][4k+2h], A[m][4k+2h+1]
        v2f b0 = *(const v2f*)(brow0 + 4 * k);  // W[o][4k+2h], o = m
        v2f b1 = *(const v2f*)(brow1 + 4 * k);  // W[o+16][...]
        acc0 = __builtin_amdgcn_wmma_f32_16x16x4_f32(
                   false, a, false, b0, (short)0, acc0, false, false);
        acc1 = __builtin_amdgcn_wmma_f32_16x16x4_f32(
                   false, a, false, b1, (short)0, acc1, false, false);
    }

    // ---------------- Store: C VGPR r holds out[face_base + r + 8h][m (+16)] ----------
    #pragma unroll
    for (int r = 0; r < 8; ++r) {
        int face = face_base + r + 8 * h;
        if (face < nfaces) {
            out[(size_t)face * 32 + m]      = acc0[r];
            out[(size_t)face * 32 + m + 16] = acc1[r];
        }
    }
}

extern "C" void kernel_launch(void* const* d_in, const int* in_sizes, int n_in,
                              void* d_out, int out_size, void* d_ws, size_t ws_size,
                              hipStream_t stream) {
    const float* x    = (const float*)d_in[0];
    const float* w0   = (const float*)d_in[1];
    const float* w1   = (const float*)d_in[2];
    const float* w2   = (const float*)d_in[3];
    const float* bias = (const float*)d_in[4];
    const int*   fn   = (const int*)d_in[5];
    // d_in[6] (face_is_pad) and d_in[7] (pad_size) unneeded: padding == idx >= nfaces
    float* out = (float*)d_out;

    const int nfaces = in_sizes[0] / C_IN;
    const int blocks = (nfaces + FACES_PER_BLOCK - 1) / FACES_PER_BLOCK;

    SymmetricFaceConv_kernel<<<blocks, 256, 0, stream>>>(
        x, w0, w1, w2, bias, fn, out, nfaces);
}